// GraphConv_30442728194070
// MI455X (gfx1250) — compile-verified
//
#include <hip/hip_runtime.h>
#include <hip/hip_bf16.h>

#define N_NODES 50000
#define N_EDGES 600000
#define DIM 128

typedef float v2f __attribute__((ext_vector_type(2)));
typedef float v8f __attribute__((ext_vector_type(8)));
typedef int   v4i_gcc __attribute__((vector_size(16)));   // matches builtin param

#if defined(__gfx1250__) && __has_builtin(__builtin_amdgcn_global_load_async_to_lds_b128) && __has_builtin(__builtin_amdgcn_s_wait_asynccnt)
#define USE_ASYNC_LDS 1
#endif

// ---------------------------------------------------------------------------
// 1) zero workspace: agg[N*128] + deg_src[N] + deg_dst[N]
// ---------------------------------------------------------------------------
__global__ void zero_ws_kernel(float* ws, int total) {
    int i = blockIdx.x * blockDim.x + threadIdx.x;
    if (i < total) ws[i] = 0.0f;
}

// ---------------------------------------------------------------------------
// 2) degree counting (float atomics; converted to norms in-place later)
// ---------------------------------------------------------------------------
__global__ void count_deg_kernel(const int* __restrict__ src,
                                 const int* __restrict__ dst,
                                 float* deg_src, float* deg_dst) {
    int e = blockIdx.x * blockDim.x + threadIdx.x;
    if (e < N_EDGES) {
        atomicAdd(&deg_src[src[e]], 1.0f);
        atomicAdd(&deg_dst[dst[e]], 1.0f);
    }
}

// ---------------------------------------------------------------------------
// 3) deg -> rsqrt(max(deg,1)) in place, both arrays
// ---------------------------------------------------------------------------
__global__ void finalize_norm_kernel(float* deg_src, float* deg_dst) {
    int i = blockIdx.x * blockDim.x + threadIdx.x;
    if (i < N_NODES) {
        deg_src[i] = rsqrtf(fmaxf(deg_src[i], 1.0f));
        deg_dst[i] = rsqrtf(fmaxf(deg_dst[i], 1.0f));
    }
}

// ---------------------------------------------------------------------------
// 4) edge aggregation: agg[dst] += feat[src] * out_norm[src]
//    one wave32 per edge, each lane handles 4 contiguous floats
// ---------------------------------------------------------------------------
__global__ __launch_bounds__(256) void aggregate_kernel(
        const float* __restrict__ feat,
        const float* __restrict__ out_norm,
        const int* __restrict__ src,
        const int* __restrict__ dst,
        float* __restrict__ agg) {
    int e = blockIdx.x * 8 + (threadIdx.x >> 5);
    if (e >= N_EDGES) return;
    int lane = threadIdx.x & 31;
    int s = src[e];
    int d = dst[e];
    float nrm = out_norm[s];
    const float4 v = *(const float4*)(feat + (size_t)s * DIM + lane * 4);
    float* a = agg + (size_t)d * DIM + lane * 4;
    atomicAdd(a + 0, v.x * nrm);
    atomicAdd(a + 1, v.y * nrm);
    atomicAdd(a + 2, v.z * nrm);
    atomicAdd(a + 3, v.w * nrm);
}

// ---------------------------------------------------------------------------
// stage one 128x128 f32 weight matrix (64 KB) into LDS with the whole block.
// Uses gfx1250 async global->LDS copies (tracked by ASYNCcnt) when available.
// ---------------------------------------------------------------------------
__device__ __forceinline__ void stage_weights(float* lds, const float* __restrict__ g,
                                              int tid) {
#ifdef USE_ASYNC_LDS
    auto ldst = (__attribute__((address_space(3))) v4i_gcc*)lds;
    auto gsrc = (__attribute__((address_space(1))) v4i_gcc*)g;   // const cast-away
#pragma unroll
    for (int i = 0; i < 16; ++i) {
        const int idx = i * 256 + tid;    // index in 16-byte chunks
        __builtin_amdgcn_global_load_async_to_lds_b128(gsrc + idx, ldst + idx, 0, 0);
    }
    __builtin_amdgcn_s_wait_asynccnt(0);
#else
#pragma unroll
    for (int i = 0; i < 16; ++i) {
        const int idx = i * 256 + tid;
        ((float4*)lds)[idx] = ((const float4*)g)[idx];
    }
#endif
    __syncthreads();
}

// ---------------------------------------------------------------------------
// 5) fused dual-GEMM (WMMA f32 16x16x4, B from LDS) + bias + in_norm +
//    residual + LayerNorm + ReLU.  Block = 256 threads = 8 waves; each wave
//    computes a 16-row x 128-col slab.  Two K-passes share one 64 KB LDS
//    weight buffer.  LayerNorm stats via half-wave shuffles.
// ---------------------------------------------------------------------------
__global__ __launch_bounds__(256) void gemm_ln_kernel(
        const float* __restrict__ agg,
        const float* __restrict__ feat,
        const float* __restrict__ fc_w,   // [128,128] row-major: fc_w[k*128+n]
        const float* __restrict__ fc_b,   // [128]
        const float* __restrict__ res_w,  // [128,128]
        const float* __restrict__ in_norm,// [N]
        const float* __restrict__ ln_g,
        const float* __restrict__ ln_b,
        float* __restrict__ out) {
    __shared__ float wlds[DIM * DIM];     // 64 KB, reused for both matrices

    const int tid  = threadIdx.x;
    const int wave = tid >> 5;
    const int lane = tid & 31;
    const int half = lane >> 4;          // 0: K rows k,k+1 / M=v ; 1: K rows k+2,k+3 / M=v+8
    const int l15  = lane & 15;
    const int row_base = blockIdx.x * 128 + wave * 16;

    // A-fragment row this lane loads (M = l15 for both halves)
    int a_row = row_base + l15;
    if (a_row >= N_NODES) a_row = N_NODES - 1;   // clamp; result masked at store
    const float* aggr  = agg  + (size_t)a_row * DIM;
    const float* featr = feat + (size_t)a_row * DIM;

    v8f acc1[8];   // agg  @ fc_w  : 8 column tiles of 16
    v8f acc2[8];   // feat @ res_w
#pragma unroll
    for (int nt = 0; nt < 8; ++nt) { acc1[nt] = (v8f)0.0f; acc2[nt] = (v8f)0.0f; }

    // ---- pass 1: acc1 += agg @ fc_w (fc_w staged in LDS) ------------------
    stage_weights(wlds, fc_w, tid);
#pragma unroll 1
    for (int k = 0; k < DIM; k += 4) {
        const int ka = k + 2 * half;     // this lane's two K indices: ka, ka+1
        v2f a;
        a[0] = aggr[ka];
        a[1] = aggr[ka + 1];
#pragma unroll
        for (int nt = 0; nt < 8; ++nt) {
            const int n = nt * 16 + l15;
            v2f b;
            b[0] = wlds[ka * DIM + n];
            b[1] = wlds[(ka + 1) * DIM + n];
            acc1[nt] = __builtin_amdgcn_wmma_f32_16x16x4_f32(
                false, a, false, b, (short)0, acc1[nt], false, false);
        }
    }

    // ---- pass 2: acc2 += feat @ res_w (res_w staged in same LDS) ----------
    __syncthreads();                      // everyone done reading fc_w
    stage_weights(wlds, res_w, tid);
#pragma unroll 1
    for (int k = 0; k < DIM; k += 4) {
        const int ka = k + 2 * half;
        v2f a;
        a[0] = featr[ka];
        a[1] = featr[ka + 1];
#pragma unroll
        for (int nt = 0; nt < 8; ++nt) {
            const int n = nt * 16 + l15;
            v2f b;
            b[0] = wlds[ka * DIM + n];
            b[1] = wlds[(ka + 1) * DIM + n];
            acc2[nt] = __builtin_amdgcn_wmma_f32_16x16x4_f32(
                false, a, false, b, (short)0, acc2[nt], false, false);
        }
    }

    // ---- epilogue: rst = (acc1 + fc_b) * in_norm[row] + acc2 --------------
    float innorm[8];
#pragma unroll
    for (int v = 0; v < 8; ++v) {
        int r = row_base + v + 8 * half;
        if (r >= N_NODES) r = N_NODES - 1;
        innorm[v] = in_norm[r];
    }
    float fcb[8], lng[8], lnb[8];
#pragma unroll
    for (int nt = 0; nt < 8; ++nt) {
        const int col = nt * 16 + l15;
        fcb[nt] = fc_b[col];
        lng[nt] = ln_g[col];
        lnb[nt] = ln_b[col];
    }

    float rst[8][8];
    float s[8], ss[8];
#pragma unroll
    for (int v = 0; v < 8; ++v) { s[v] = 0.0f; ss[v] = 0.0f; }
#pragma unroll
    for (int nt = 0; nt < 8; ++nt) {
#pragma unroll
        for (int v = 0; v < 8; ++v) {
            float x = (acc1[nt][v] + fcb[nt]) * innorm[v] + acc2[nt][v];
            rst[nt][v] = x;
            s[v]  += x;
            ss[v] += x * x;
        }
    }
    // reduce across the 16-lane half (bits 0..3 only; bit4 separates row sets)
#pragma unroll
    for (int v = 0; v < 8; ++v) {
#pragma unroll
        for (int off = 1; off <= 8; off <<= 1) {
            s[v]  += __shfl_xor(s[v],  off, 32);
            ss[v] += __shfl_xor(ss[v], off, 32);
        }
    }
    float mean[8], rstd[8];
#pragma unroll
    for (int v = 0; v < 8; ++v) {
        mean[v] = s[v] * (1.0f / DIM);
        float var = ss[v] * (1.0f / DIM) - mean[v] * mean[v];
        rstd[v] = rsqrtf(var + 1e-5f);
    }
    // ---- LayerNorm + ReLU + store ----------------------------------------
#pragma unroll
    for (int v = 0; v < 8; ++v) {
        const int row = row_base + v + 8 * half;
        if (row < N_NODES) {
#pragma unroll
            for (int nt = 0; nt < 8; ++nt) {
                const int col = nt * 16 + l15;
                float y = (rst[nt][v] - mean[v]) * rstd[v] * lng[nt] + lnb[nt];
                out[(size_t)row * DIM + col] = fmaxf(y, 0.0f);
            }
        }
    }
}

// ---------------------------------------------------------------------------
extern "C" void kernel_launch(void* const* d_in, const int* in_sizes, int n_in,
                              void* d_out, int out_size, void* d_ws, size_t ws_size,
                              hipStream_t stream) {
    const float* feat  = (const float*)d_in[0];
    const int*   src   = (const int*)  d_in[1];
    const int*   dst   = (const int*)  d_in[2];
    const float* fc_w  = (const float*)d_in[3];
    const float* fc_b  = (const float*)d_in[4];
    const float* res_w = (const float*)d_in[5];
    const float* ln_g  = (const float*)d_in[6];
    const float* ln_b  = (const float*)d_in[7];
    float* out = (float*)d_out;

    // workspace layout
    float* agg      = (float*)d_ws;                 // N*128
    float* deg_src  = agg + (size_t)N_NODES * DIM;  // N  (becomes out_norm)
    float* deg_dst  = deg_src + N_NODES;            // N  (becomes in_norm)

    const int total_zero = N_NODES * DIM + 2 * N_NODES;
    zero_ws_kernel<<<(total_zero + 255) / 256, 256, 0, stream>>>(agg, total_zero);

    count_deg_kernel<<<(N_EDGES + 255) / 256, 256, 0, stream>>>(src, dst, deg_src, deg_dst);

    finalize_norm_kernel<<<(N_NODES + 255) / 256, 256, 0, stream>>>(deg_src, deg_dst);

    aggregate_kernel<<<(N_EDGES + 7) / 8, 256, 0, stream>>>(feat, deg_src, src, dst, agg);

    const int row_blocks = (N_NODES + 127) / 128;   // 391
    gemm_ln_kernel<<<row_blocks, 256, 0, stream>>>(
        agg, feat, fc_w, fc_b, res_w, deg_dst, ln_g, ln_b, out);
}